// PointVoxelCNN_80221399154837
// MI455X (gfx1250) — compile-verified
//
#include <hip/hip_runtime.h>

// Problem constants (from reference)
#define BB   4
#define NN   65536
#define CINN 32
#define CCH  64
#define RRES 32
#define RV3  32768
#define NGRP 32

typedef float v2f __attribute__((ext_vector_type(2)));
typedef float v8f __attribute__((ext_vector_type(8)));
typedef int   v4i __attribute__((ext_vector_type(4)));

#if defined(__gfx1250__) && __has_builtin(__builtin_amdgcn_global_load_async_to_lds_b128)
#define HAS_ASYNC_LDS 1
#else
#define HAS_ASYNC_LDS 0
#endif

static __device__ __forceinline__ v8f wmma4(v2f a, v2f b, v8f c) {
  // D = A(16x4) x B(4x16) + C, fp32
  return __builtin_amdgcn_wmma_f32_16x16x4_f32(false, a, false, b, (short)0, c, false, false);
}

static __device__ __forceinline__ float leaky(float x) { return x > 0.f ? x : 0.01f * x; }

template<int N>
static __device__ __forceinline__ void wait_async() {
#if defined(__gfx1250__)
#if __has_builtin(__builtin_amdgcn_s_wait_asynccnt)
  __builtin_amdgcn_s_wait_asynccnt(N);
#else
  asm volatile("s_wait_asynccnt %0" :: "i"(N) : "memory");
#endif
#endif
}

// Stage CIN_T/16 x 16 bytes/thread from tap-major weights into LDS asynchronously.
template<int CIN_T>
static __device__ __forceinline__ void stage_async(const float* __restrict__ src,
                                                   float* dst, int tid) {
#if HAS_ASYNC_LDS
  constexpr int NCH = CIN_T / 16;   // b128 chunks per thread (256 thr, 64*CIN_T floats)
#pragma unroll
  for (int j = 0; j < NCH; ++j) {
    const int e = (tid + j * 256) * 4;  // float index of 16B chunk
    float* s = const_cast<float*>(src + e);
    __builtin_amdgcn_global_load_async_to_lds_b128(
        (__attribute__((address_space(1))) v4i*)s,
        (__attribute__((address_space(3))) v4i*)(dst + e),
        0, 0);
  }
#endif
}

// ---------------------------------------------------------------------------
// GroupNorm statistics: one block per (b, g); input ch-first [B, 64, M].
// stats[(b*G+g)*2] = mean, [..+1] = rsqrt(var+eps).
// ---------------------------------------------------------------------------
__global__ void __launch_bounds__(256) gn_stats_kernel(const float* __restrict__ x,
                                                       int M, float* __restrict__ stats) {
  const int tid = threadIdx.x;
  const int b = blockIdx.x >> 5;
  const int g = blockIdx.x & 31;
  const float* p = x + ((size_t)(b * CCH + g * 2)) * (size_t)M; // 2 contiguous channels
  const int tot = 2 * M;
  float s = 0.f, s2 = 0.f;
  for (int i = tid; i < tot; i += 256) { float v = p[i]; s += v; s2 += v * v; }
  __shared__ float rs[256], rq[256];
  rs[tid] = s; rq[tid] = s2;
  __syncthreads();
  for (int off = 128; off > 0; off >>= 1) {
    if (tid < off) { rs[tid] += rs[tid + off]; rq[tid] += rq[tid + off]; }
    __syncthreads();
  }
  if (tid == 0) {
    float inv = 1.f / (float)tot;
    float mean = rs[0] * inv;
    float var  = rq[0] * inv - mean * mean;
    stats[blockIdx.x * 2 + 0] = mean;
    stats[blockIdx.x * 2 + 1] = rsqrtf(var + 1e-5f);
  }
}

// ---------------------------------------------------------------------------
// Point-branch 1x1 conv GEMM on WMMA f32 16x16x4.
// ACT=false: input = features channel-last [B, N, KDIM]; no activation.
// ACT=true : input channel-first [B, KDIM, N]; leaky(gn(.)) fused on load.
// FINAL    : out[b,n,c] = (conv + resid)/2  (== ((h+fp)/sqrt2)/sqrt2)
// ---------------------------------------------------------------------------
template<int KDIM, bool ACT, bool FINAL>
__global__ void __launch_bounds__(256) pt_gemm_kernel(
    const float* __restrict__ in, const float* __restrict__ W, const float* __restrict__ bias,
    const float* __restrict__ gamma, const float* __restrict__ beta,
    const float* __restrict__ stats, const float* __restrict__ resid,
    float* __restrict__ out)
{
  __shared__ float sW[CCH * KDIM];
  __shared__ float sScale[KDIM];
  __shared__ float sShift[KDIM];
  const int tid  = threadIdx.x;
  const int wave = tid >> 5, lane = tid & 31;
  const int b    = blockIdx.x >> 9;                       // 512 blocks per batch
  const int tile = ((blockIdx.x & 511) << 3) | wave;      // 16-point tile within batch
  const int n0   = tile << 4;

  for (int i = tid; i < CCH * KDIM; i += 256) sW[i] = W[i];
  if (ACT) {
    for (int ch = tid; ch < KDIM; ch += 256) {
      const int g = ch >> 1;
      const float mean = stats[(b * NGRP + g) * 2 + 0];
      const float rstd = stats[(b * NGRP + g) * 2 + 1];
      const float sc = gamma[ch] * rstd;
      sScale[ch] = sc;
      sShift[ch] = beta[ch] - mean * sc;
    }
  }
  __syncthreads();

  const int nl    = lane & 15;          // N (column) index within tile
  const int krow2 = (lane >> 4) << 1;   // K pair base (0 or 2)
  const int mrow  = (lane >> 4) << 3;   // C/D: high lanes carry M+8

  v8f acc[4] = {};
  for (int k0 = 0; k0 < KDIM; k0 += 4) {
    const int c0 = k0 + krow2;
    float x0, x1;
    if (ACT) { // channel-first [B, KDIM, N]
      const float* p = in + ((size_t)(b * KDIM + c0)) * NN + n0 + nl;
      x0 = p[0]; x1 = p[NN];
      x0 = leaky(fmaf(x0, sScale[c0],     sShift[c0]));
      x1 = leaky(fmaf(x1, sScale[c0 + 1], sShift[c0 + 1]));
    } else {   // channel-last features [B, N, KDIM]
      const float* p = in + ((size_t)b * NN + n0 + nl) * KDIM + c0;
      x0 = p[0]; x1 = p[1];
    }
    v2f bf; bf.x = x0; bf.y = x1;
#pragma unroll
    for (int t = 0; t < 4; ++t) {
      const float* wr = &sW[(t * 16 + nl) * KDIM + c0];
      v2f af; af.x = wr[0]; af.y = wr[1];
      acc[t] = wmma4(af, bf, acc[t]);
    }
  }

#pragma unroll
  for (int t = 0; t < 4; ++t) {
#pragma unroll
    for (int r = 0; r < 8; ++r) {
      const int co = t * 16 + mrow + r;
      float v = acc[t][r] + bias[co];
      if (FINAL) {
        const float f = resid[((size_t)(b * CCH + co)) * NN + n0 + nl];
        out[((size_t)b * NN + n0 + nl) * CCH + co] = (v + f) * 0.5f;
      } else {
        out[((size_t)(b * CCH + co)) * NN + n0 + nl] = v;
      }
    }
  }
}

// ---------------------------------------------------------------------------
// Voxelize: nearest-voxel scatter-add of features + counts (atomics).
// ---------------------------------------------------------------------------
__global__ void __launch_bounds__(256) voxelize_kernel(
    const float* __restrict__ points, const float* __restrict__ feats,
    float* __restrict__ sums, float* __restrict__ cnt)
{
  const size_t t = (size_t)blockIdx.x * 256 + threadIdx.x;
  const size_t p = t >> 5; const int ci = (int)(t & 31);
  const float* pp = points + p * 3;
  const int ix = min(max((int)floorf((pp[0] * 0.5f + 0.5f) * 32.f), 0), 31);
  const int iy = min(max((int)floorf((pp[1] * 0.5f + 0.5f) * 32.f), 0), 31);
  const int iz = min(max((int)floorf((pp[2] * 0.5f + 0.5f) * 32.f), 0), 31);
  const size_t b = p >> 16; // N = 65536
  const size_t flat = (size_t)((ix * 32 + iy) * 32 + iz) + b * RV3;
  atomicAdd(&sums[flat * CINN + ci], feats[p * CINN + ci]);
  if (ci == 0) atomicAdd(&cnt[flat], 1.0f);
}

__global__ void __launch_bounds__(256) vox_norm_kernel(float* __restrict__ sums,
                                                       const float* __restrict__ cnt) {
  const size_t t = (size_t)blockIdx.x * 256 + threadIdx.x;
  const float c = cnt[t >> 5];
  sums[t] = sums[t] / fmaxf(c, 1.0f);
}

// ---------------------------------------------------------------------------
// Weight transpose: [C, CIN_T, 27] (tap innermost) -> [27][C*CIN_T] tap-major,
// so each tap slice is contiguous (async b128 friendly).
// ---------------------------------------------------------------------------
template<int CIN_T>
__global__ void __launch_bounds__(256) wt_transpose_kernel(const float* __restrict__ W,
                                                           float* __restrict__ Wt) {
  const int t = blockIdx.x * 256 + threadIdx.x;   // over 27 * 64*CIN_T, exact grid
  const int tap = t / (CCH * CIN_T);
  const int i   = t - tap * (CCH * CIN_T);
  Wt[(size_t)tap * (CCH * CIN_T) + i] = W[(size_t)i * 27 + tap];
}

// ---------------------------------------------------------------------------
// 3x3x3 conv as implicit GEMM on WMMA f32 16x16x4.
// Weights come in tap-major layout Wt[27][64*CIN_T]; per-tap slice is staged
// into LDS with double-buffered async global->LDS copies (ASYNCcnt) when the
// toolchain exposes them, else coalesced sync staging.
// CIN_T==32: input channel-last [B*R3, 32] (normalized grid0).
// CIN_T==64: input channel-first [B, 64, R3]; ACT fuses leaky(gn(.)).
// FINAL: out_grid[b, vox, co] = (conv + input_grid)/sqrt2 (channel-last).
// ---------------------------------------------------------------------------
template<int CIN_T, bool ACT, bool FINAL>
__global__ void __launch_bounds__(256) conv3d_kernel(
    const float* __restrict__ Wt, const float* __restrict__ in, const float* __restrict__ bias,
    const float* __restrict__ gamma, const float* __restrict__ beta,
    const float* __restrict__ stats, const float* __restrict__ resid,
    float* __restrict__ out)
{
#if HAS_ASYNC_LDS
  __shared__ float sW[2][CCH * CIN_T];
#else
  __shared__ float sW[1][CCH * CIN_T];
#endif
  __shared__ float sScale[CIN_T];
  __shared__ float sShift[CIN_T];
  const int tid  = threadIdx.x;
  const int wave = tid >> 5, lane = tid & 31;
  const int b   = blockIdx.x >> 8;          // 32 z * 8 y-groups = 256 blocks/batch
  const int rem = blockIdx.x & 255;
  const int z   = rem >> 3;
  const int y   = ((rem & 7) << 2) + (wave >> 1);
  const int x0  = (wave & 1) << 4;

  if (ACT) {
    for (int ch = tid; ch < CIN_T; ch += 256) {
      const int g = ch >> 1;
      const float mean = stats[(b * NGRP + g) * 2 + 0];
      const float rstd = stats[(b * NGRP + g) * 2 + 1];
      const float sc = gamma[ch] * rstd;
      sScale[ch] = sc;
      sShift[ch] = beta[ch] - mean * sc;
    }
  }

  const int nl    = lane & 15;
  const int krow2 = (lane >> 4) << 1;
  const int mrow  = (lane >> 4) << 3;

#if HAS_ASYNC_LDS
  stage_async<CIN_T>(Wt, sW[0], tid);       // prologue: tap 0 in flight
#endif

  v8f acc[4] = {};
  for (int tap = 0; tap < 27; ++tap) {
    const int dz = tap / 9, dy = (tap / 3) % 3, dx = tap % 3;
#if HAS_ASYNC_LDS
    if (tap + 1 < 27)
      stage_async<CIN_T>(Wt + (size_t)(tap + 1) * (CCH * CIN_T), sW[(tap + 1) & 1], tid);
    if (tap + 1 < 27) wait_async<CIN_T / 16>(); else wait_async<0>();
    __syncthreads();                         // this tap's slice visible block-wide
    const float* sWc = sW[tap & 1];
#else
    __syncthreads();
    for (int i = tid; i < CCH * CIN_T; i += 256)
      sW[0][i] = Wt[(size_t)tap * (CCH * CIN_T) + i];
    __syncthreads();
    const float* sWc = sW[0];
#endif
    const int zi = z + dz - 1, yi = y + dy - 1;
    if ((unsigned)zi < 32u && (unsigned)yi < 32u) {   // wave-uniform halo skip
      const int xl = x0 + nl + dx - 1;
      const bool xok = (unsigned)xl < 32u;
      const size_t vox = (size_t)((zi * 32 + yi) * 32 + (xok ? xl : 0));
      for (int k0 = 0; k0 < CIN_T; k0 += 4) {
        const int c0 = k0 + krow2;
        float x0v, x1v;
        if (CIN_T == 32) {
          const float* p = in + ((size_t)b * RV3 + vox) * 32 + c0;
          x0v = p[0]; x1v = p[1];
        } else {
          const float* p = in + ((size_t)(b * CCH + c0)) * RV3 + vox;
          x0v = p[0]; x1v = p[RV3];
        }
        if (ACT) {
          x0v = leaky(fmaf(x0v, sScale[c0],     sShift[c0]));
          x1v = leaky(fmaf(x1v, sScale[c0 + 1], sShift[c0 + 1]));
        }
        if (!xok) { x0v = 0.f; x1v = 0.f; }
        v2f bf; bf.x = x0v; bf.y = x1v;
#pragma unroll
        for (int t = 0; t < 4; ++t) {
          const float* wr = &sWc[(t * 16 + nl) * CIN_T + c0];
          v2f af; af.x = wr[0]; af.y = wr[1];
          acc[t] = wmma4(af, bf, acc[t]);
        }
      }
    }
#if HAS_ASYNC_LDS
    __syncthreads();   // protect buffer from tap+2 overwrite
#endif
  }

  const size_t vox0 = (size_t)((z * 32 + y) * 32 + x0 + nl);
#pragma unroll
  for (int t = 0; t < 4; ++t) {
#pragma unroll
    for (int r = 0; r < 8; ++r) {
      const int co = t * 16 + mrow + r;
      float v = acc[t][r] + bias[co];
      if (FINAL) {
        const float ig = resid[((size_t)(b * CCH + co)) * RV3 + vox0];
        out[((size_t)b * RV3 + vox0) * CCH + co] = (v + ig) * 0.70710678118654752f;
      } else {
        out[((size_t)(b * CCH + co)) * RV3 + vox0] = v;
      }
    }
  }
}

// ---------------------------------------------------------------------------
// Trilinear devoxelize-gather; adds vx/sqrt2 into d_out (point branch wrote it).
// ---------------------------------------------------------------------------
__global__ void __launch_bounds__(256) devox_kernel(
    const float* __restrict__ points, const float* __restrict__ og, float* __restrict__ outp)
{
  const size_t t = (size_t)blockIdx.x * 256 + threadIdx.x;
  const size_t p = t >> 6; const int ch = (int)(t & 63);
  const float* pp = points + p * 3;
  const size_t b = p >> 16;
  const float c0 = fminf(fmaxf((pp[0] * 0.5f + 0.5f) * 31.f, 0.f), 31.f);
  const float c1 = fminf(fmaxf((pp[1] * 0.5f + 0.5f) * 31.f, 0.f), 31.f);
  const float c2 = fminf(fmaxf((pp[2] * 0.5f + 0.5f) * 31.f, 0.f), 31.f);
  const int l0 = (int)floorf(c0), l1 = (int)floorf(c1), l2 = (int)floorf(c2);
  const float f0 = c0 - (float)l0, f1 = c1 - (float)l1, f2 = c2 - (float)l2;
  const int h0 = min(l0 + 1, 31), h1 = min(l1 + 1, 31), h2 = min(l2 + 1, 31);
  float a = 0.f;
#pragma unroll
  for (int d = 0; d < 8; ++d) {
    const int   i0 = (d & 4) ? h0 : l0; const float w0 = (d & 4) ? f0 : 1.f - f0;
    const int   i1 = (d & 2) ? h1 : l1; const float w1 = (d & 2) ? f1 : 1.f - f1;
    const int   i2 = (d & 1) ? h2 : l2; const float w2 = (d & 1) ? f2 : 1.f - f2;
    const size_t fl = (size_t)((i0 * 32 + i1) * 32 + i2) + b * RV3;
    a += (w0 * w1 * w2) * og[fl * CCH + ch];
  }
  outp[p * CCH + ch] += a * 0.70710678118654752f;
}

// ---------------------------------------------------------------------------
extern "C" void kernel_launch(void* const* d_in, const int* in_sizes, int n_in,
                              void* d_out, int out_size, void* d_ws, size_t ws_size,
                              hipStream_t stream) {
  (void)in_sizes; (void)n_in; (void)out_size; (void)ws_size;
  const float* points   = (const float*)d_in[0];
  const float* features = (const float*)d_in[1];
  const float* w_in3d = (const float*)d_in[2];
  const float* b_in3d = (const float*)d_in[3];
  const float* w_c1   = (const float*)d_in[4];
  const float* b_c1   = (const float*)d_in[5];
  const float* w_c2   = (const float*)d_in[6];
  const float* b_c2   = (const float*)d_in[7];
  const float* g1  = (const float*)d_in[8];
  const float* be1 = (const float*)d_in[9];
  const float* g2  = (const float*)d_in[10];
  const float* be2 = (const float*)d_in[11];
  const float* w_pin = (const float*)d_in[12];
  const float* b_pin = (const float*)d_in[13];
  const float* w_pc1 = (const float*)d_in[14];
  const float* b_pc1 = (const float*)d_in[15];
  const float* w_pc2 = (const float*)d_in[16];
  const float* b_pc2 = (const float*)d_in[17];
  const float* pg1 = (const float*)d_in[18];
  const float* pb1 = (const float*)d_in[19];
  const float* pg2 = (const float*)d_in[20];
  const float* pb2 = (const float*)d_in[21];
  float* out = (float*)d_out;
  char* ws = (char*)d_ws;

  const size_t SZ_FP = (size_t)BB * CCH * NN * sizeof(float); // 64 MB
  float* fp = (float*)ws;
  float* h1 = (float*)(ws + SZ_FP);
  float* statsA = (float*)(ws + 2 * SZ_FP);                   // 2 x 256 floats
  float* statsB = statsA + 2 * BB * NGRP;

  // ---- point branch ----
  pt_gemm_kernel<CINN, false, false><<<2048, 256, 0, stream>>>(
      features, w_pin, b_pin, nullptr, nullptr, nullptr, nullptr, fp);
  gn_stats_kernel<<<BB * NGRP, 256, 0, stream>>>(fp, NN, statsA);
  pt_gemm_kernel<CCH, true, false><<<2048, 256, 0, stream>>>(
      fp, w_pc1, b_pc1, pg1, pb1, statsA, nullptr, h1);
  gn_stats_kernel<<<BB * NGRP, 256, 0, stream>>>(h1, NN, statsB);
  pt_gemm_kernel<CCH, true, true><<<2048, 256, 0, stream>>>(
      h1, w_pc2, b_pc2, pg2, pb2, statsB, fp, out);           // out = (h+fp)/2

  // ---- voxel branch (aliases the fp/h1 region; point branch is done) ----
  float* sums = (float*)ws;                                   // [B*R3, 32]  16 MB
  float* cnt  = (float*)(ws + 16777216);                      // [B*R3]      0.5 MB
  float* ig   = (float*)(ws + 16777216 + 524288);             // [B,64,R3]   32 MB
  float* gg1  = (float*)(ws + 16777216 + 524288 + 33554432);  // [B,64,R3]   32 MB
  float* og   = (float*)(ws + 16777216 + 524288 + 67108864);  // [B*R3,64]   32 MB
  float* Wt   = (float*)(ws + 16777216 + 524288 + 100663296); // 27*64*64*4 = 432 KB

  (void)hipMemsetAsync(sums, 0, (size_t)BB * RV3 * CINN * sizeof(float), stream);
  (void)hipMemsetAsync(cnt,  0, (size_t)BB * RV3 * sizeof(float), stream);
  voxelize_kernel<<<(BB * NN * CINN) / 256, 256, 0, stream>>>(points, features, sums, cnt);
  vox_norm_kernel<<<(BB * RV3 * CINN) / 256, 256, 0, stream>>>(sums, cnt);

  wt_transpose_kernel<CINN><<<(27 * CCH * CINN) / 256, 256, 0, stream>>>(w_in3d, Wt);
  conv3d_kernel<CINN, false, false><<<BB * 256, 256, 0, stream>>>(
      Wt, sums, b_in3d, nullptr, nullptr, nullptr, nullptr, ig);
  gn_stats_kernel<<<BB * NGRP, 256, 0, stream>>>(ig, RV3, statsA);

  wt_transpose_kernel<CCH><<<(27 * CCH * CCH) / 256, 256, 0, stream>>>(w_c1, Wt);
  conv3d_kernel<CCH, true, false><<<BB * 256, 256, 0, stream>>>(
      Wt, ig, b_c1, g1, be1, statsA, nullptr, gg1);
  gn_stats_kernel<<<BB * NGRP, 256, 0, stream>>>(gg1, RV3, statsB);

  wt_transpose_kernel<CCH><<<(27 * CCH * CCH) / 256, 256, 0, stream>>>(w_c2, Wt);
  conv3d_kernel<CCH, true, true><<<BB * 256, 256, 0, stream>>>(
      Wt, gg1, b_c2, g2, be2, statsB, ig, og);                // out_grid = (gg+ig)/sqrt2
  devox_kernel<<<(BB * NN * CCH) / 256, 256, 0, stream>>>(points, og, out);
}